// GatMeanPool_4355096838777
// MI455X (gfx1250) — compile-verified
//
#include <hip/hip_runtime.h>
#include <hip/hip_bf16.h>
#include <cstdint>

#define DHID 128

typedef float v8f __attribute__((ext_vector_type(8)));
typedef float v2f __attribute__((ext_vector_type(2)));

// ---------------- order-preserving float <-> u32 key (for atomic segment-max) ---
__device__ __forceinline__ unsigned fkey(float f) {
    unsigned u = __float_as_uint(f);
    return (u & 0x80000000u) ? ~u : (u | 0x80000000u);
}
__device__ __forceinline__ float funkey(unsigned k) {
    unsigned u = (k & 0x80000000u) ? (k & 0x7fffffffu) : ~k;
    return __uint_as_float(u);
}

// ---------------- generic u32 fill (graph-capture-safe init) --------------------
__global__ void fill_u32_kernel(unsigned* __restrict__ p, unsigned v, long n) {
    long i = (long)blockIdx.x * blockDim.x + threadIdx.x;
    if (i < n) p[i] = v;
}

// ---------------- WMMA fp32 GEMM: H[rows,128] = X[rows,128] @ W[128,128] (+bias)
// Block: 256 threads = 8 waves; block computes a 16-row x 128-col tile.
// Wave w owns the 16x16 tile at columns [16w,16w+16), sweeping K=128 via 32x
// v_wmma_f32_16x16x4_f32. X tile staged in LDS, 132-float pitch (132%64==4 ->
// the 16 A-fragment lanes map to 16 distinct banks). Epilogue uses a single
// uniform full-tile check so the common path is 8 unguarded coalesced stores.
__global__ __launch_bounds__(256) void wmma_gemm128(
    const float* __restrict__ X, const float* __restrict__ W,
    const float* __restrict__ bias, float* __restrict__ H, int nrows)
{
    __shared__ float sX[16 * 132];
    const int tid  = threadIdx.x;
    const int row0 = blockIdx.x * 16;

    for (int i = tid; i < 16 * 32; i += 256) {
        int r = i >> 5, c4 = i & 31;
        int gr = row0 + r;
        float4 v = make_float4(0.f, 0.f, 0.f, 0.f);
        if (gr < nrows) v = ((const float4*)X)[(size_t)gr * 32 + c4];
        *(float4*)(&sX[r * 132 + c4 * 4]) = v;
    }
    __syncthreads();

    const int wave  = tid >> 5;
    const int lane  = tid & 31;
    const int n0    = wave * 16;
    const int l16   = lane & 15;
    const int khalf = lane >> 4;     // A/B K-pair select: {k,k+1} or {k+2,k+3}

    v8f acc = {};
    #pragma unroll 4
    for (int k0 = 0; k0 < DHID; k0 += 4) {
        const int ka = k0 + 2 * khalf;
        v2f a, b;
        a.x = sX[l16 * 132 + ka];
        a.y = sX[l16 * 132 + ka + 1];
        const int col = n0 + l16;
        b.x = W[(size_t)ka * DHID + col];
        b.y = W[(size_t)(ka + 1) * DHID + col];
        acc = __builtin_amdgcn_wmma_f32_16x16x4_f32(
            false, a, false, b, (short)0, acc, false, false);
    }

    // C/D layout: vgpr v, lanes 0-15 -> M=v, lanes 16-31 -> M=v+8; N = l16
    const int nc    = n0 + l16;
    const int mbase = row0 + 8 * khalf;
    const float bb  = bias ? bias[nc] : 0.f;
    if (row0 + 16 <= nrows) {
        // full tile: unconditional, per-row 64B-coalesced stores
        #pragma unroll
        for (int v = 0; v < 8; ++v)
            H[(size_t)(mbase + v) * DHID + nc] = acc[v] + bb;
    } else {
        #pragma unroll
        for (int v = 0; v < 8; ++v) {
            int gr = mbase + v;
            if (gr < nrows) H[(size_t)gr * DHID + nc] = acc[v] + bb;
        }
    }
}

// ---------------- per-node attention dots: as = H.a_src, ad = H.a_dst ----------
__global__ __launch_bounds__(256) void node_dots(
    const float* __restrict__ H, const float* __restrict__ a_src,
    const float* __restrict__ a_dst, float* __restrict__ as_,
    float* __restrict__ ad_, int n)
{
    int row  = blockIdx.x * (blockDim.x >> 5) + (threadIdx.x >> 5);
    int lane = threadIdx.x & 31;
    if (row >= n) return;
    float4 hv = ((const float4*)(H + (size_t)row * DHID))[lane];
    float4 sv = ((const float4*)a_src)[lane];
    float4 dv = ((const float4*)a_dst)[lane];
    float accs = hv.x * sv.x + hv.y * sv.y + hv.z * sv.z + hv.w * sv.w;
    float accd = hv.x * dv.x + hv.y * dv.y + hv.z * dv.z + hv.w * dv.w;
    #pragma unroll
    for (int off = 16; off > 0; off >>= 1) {
        accs += __shfl_xor(accs, off, 32);
        accd += __shfl_xor(accd, off, 32);
    }
    if (lane == 0) { as_[row] = accs; ad_[row] = accd; }
}

// ---------------- edge logits + atomic segment-max over dst --------------------
__global__ __launch_bounds__(256) void edge_logits(
    const int* __restrict__ ei, int E_, int n,
    const float* __restrict__ as_, const float* __restrict__ ad_,
    float* __restrict__ logit, unsigned* __restrict__ mkey)
{
    int e = blockIdx.x * blockDim.x + threadIdx.x;
    int total = E_ + n;
    if (e >= total) return;
    int s, d;
    if (e < E_) { s = ei[e]; d = ei[E_ + e]; } else { s = d = e - E_; }  // self-loop
    float l = as_[s] + ad_[d];
    l = (l > 0.f) ? l : 0.2f * l;            // leaky_relu(0.2)
    logit[e] = l;
    atomicMax(mkey + d, fkey(l));
}

// ---------------- exp(logit - max) + atomic denominator ------------------------
__global__ __launch_bounds__(256) void edge_exp(
    const int* __restrict__ ei, int E_, int n,
    float* __restrict__ pe, const unsigned* __restrict__ mkey,
    float* __restrict__ denom)
{
    int e = blockIdx.x * blockDim.x + threadIdx.x;
    int total = E_ + n;
    if (e >= total) return;
    int d = (e < E_) ? ei[E_ + e] : (e - E_);
    float p = __expf(pe[e] - funkey(mkey[d]));
    pe[e] = p;
    atomicAdd(denom + d, p);
}

// ---------------- weighted scatter-add: out[dst] += alpha * h[src] -------------
__global__ __launch_bounds__(256) void edge_aggregate(
    const int* __restrict__ ei, int E_, int n,
    const float* __restrict__ pe, const float* __restrict__ denom,
    const float* __restrict__ H, float* __restrict__ out)
{
    int wid  = (int)(((long)blockIdx.x * blockDim.x + threadIdx.x) >> 5);
    int lane = threadIdx.x & 31;
    int total = E_ + n;
    if (wid >= total) return;
    int s, d;
    if (wid < E_) { s = ei[wid]; d = ei[E_ + wid]; } else { s = d = wid - E_; }
    float alpha = pe[wid] / (denom[d] + 1e-16f);
    float4 hv = ((const float4*)(H + (size_t)s * DHID))[lane];
    float* o = out + (size_t)d * DHID + lane * 4;
    atomicAdd(o + 0, alpha * hv.x);
    atomicAdd(o + 1, alpha * hv.y);
    atomicAdd(o + 2, alpha * hv.z);
    atomicAdd(o + 3, alpha * hv.w);
}

// ---------------- bias (+optional ReLU) ----------------------------------------
__global__ __launch_bounds__(256) void bias_act(
    const float* __restrict__ agg, const float* __restrict__ b,
    float* __restrict__ out, long total, int relu)
{
    long i = (long)blockIdx.x * blockDim.x + threadIdx.x;
    if (i >= total) return;
    float v = agg[i] + b[i & (DHID - 1)];
    if (relu) v = fmaxf(v, 0.f);
    out[i] = v;
}

// ---------------- global mean pool ---------------------------------------------
__global__ __launch_bounds__(256) void pool_accum(
    const float* __restrict__ h, const int* __restrict__ batch,
    float* __restrict__ sums, float* __restrict__ cnts, int n)
{
    int row  = blockIdx.x * (blockDim.x >> 5) + (threadIdx.x >> 5);
    int lane = threadIdx.x & 31;
    if (row >= n) return;
    int g = batch[row];
    float4 v = ((const float4*)(h + (size_t)row * DHID))[lane];
    float* s = sums + (size_t)g * DHID + lane * 4;
    atomicAdd(s + 0, v.x); atomicAdd(s + 1, v.y);
    atomicAdd(s + 2, v.z); atomicAdd(s + 3, v.w);
    if (lane == 0) atomicAdd(cnts + g, 1.0f);
}

__global__ __launch_bounds__(256) void pool_div(
    const float* __restrict__ sums, const float* __restrict__ cnts,
    float* __restrict__ pooled, int Gn)
{
    int i = blockIdx.x * blockDim.x + threadIdx.x;
    if (i >= Gn * DHID) return;
    pooled[i] = sums[i] / fmaxf(cnts[i >> 7], 1.0f);
}

// ================================================================================
extern "C" void kernel_launch(void* const* d_in, const int* in_sizes, int n_in,
                              void* d_out, int out_size, void* d_ws, size_t ws_size,
                              hipStream_t stream) {
    const float* x     = (const float*)d_in[0];
    const int*   ei    = (const int*)d_in[1];     // [2, E] row-major
    const int*   batch = (const int*)d_in[3];
    const float* Wg[3]   = {(const float*)d_in[4],  (const float*)d_in[8],  (const float*)d_in[12]};
    const float* Asrc[3] = {(const float*)d_in[5],  (const float*)d_in[9],  (const float*)d_in[13]};
    const float* Adst[3] = {(const float*)d_in[6],  (const float*)d_in[10], (const float*)d_in[14]};
    const float* Bias[3] = {(const float*)d_in[7],  (const float*)d_in[11], (const float*)d_in[15]};
    const float* linW = (const float*)d_in[16];
    const float* linB = (const float*)d_in[17];

    const int N    = in_sizes[0] / DHID;
    const int E    = in_sizes[1] / 2;
    const int Etot = E + N;                       // with self-loops
    const int Gn   = out_size / DHID;

    char* ws = (char*)d_ws;
    size_t off = 0;
    auto carve = [&](size_t bytes) -> char* {
        char* p = ws + off;
        off += (bytes + 255) & ~(size_t)255;
        return p;
    };
    float*    hw     = (float*)carve((size_t)N * DHID * 4);
    float*    agg    = (float*)carve((size_t)N * DHID * 4);
    float*    xin    = (float*)carve((size_t)N * DHID * 4);
    float*    as_    = (float*)carve((size_t)N * 4);
    float*    ad_    = (float*)carve((size_t)N * 4);
    unsigned* mkey   = (unsigned*)carve((size_t)N * 4);
    float*    denom  = (float*)carve((size_t)N * 4);
    float*    pe     = (float*)carve((size_t)Etot * 4);
    float*    sums   = (float*)carve((size_t)Gn * DHID * 4);
    float*    cnts   = (float*)carve((size_t)Gn * 4);
    float*    pooled = (float*)carve((size_t)Gn * DHID * 4);
    (void)ws_size; (void)n_in;

    const long nd = (long)N * DHID;
    const float* cur = x;
    for (int l = 0; l < 3; ++l) {
        wmma_gemm128<<<(N + 15) / 16, 256, 0, stream>>>(cur, Wg[l], nullptr, hw, N);
        node_dots<<<(N + 7) / 8, 256, 0, stream>>>(hw, Asrc[l], Adst[l], as_, ad_, N);
        fill_u32_kernel<<<(N + 255) / 256, 256, 0, stream>>>(mkey, 0x007FFFFFu /*fkey(-inf)*/, N);
        fill_u32_kernel<<<(N + 255) / 256, 256, 0, stream>>>((unsigned*)denom, 0u, N);
        fill_u32_kernel<<<(unsigned)((nd + 255) / 256), 256, 0, stream>>>((unsigned*)agg, 0u, nd);
        edge_logits<<<(Etot + 255) / 256, 256, 0, stream>>>(ei, E, N, as_, ad_, pe, mkey);
        edge_exp<<<(Etot + 255) / 256, 256, 0, stream>>>(ei, E, N, pe, mkey, denom);
        edge_aggregate<<<(Etot + 7) / 8, 256, 0, stream>>>(ei, E, N, pe, denom, hw, agg);
        bias_act<<<(unsigned)((nd + 255) / 256), 256, 0, stream>>>(agg, Bias[l], xin, nd, l < 2);
        cur = xin;
    }

    const long gd = (long)Gn * DHID;
    fill_u32_kernel<<<(unsigned)((gd + 255) / 256), 256, 0, stream>>>((unsigned*)sums, 0u, gd);
    fill_u32_kernel<<<(Gn + 255) / 256, 256, 0, stream>>>((unsigned*)cnts, 0u, Gn);
    pool_accum<<<(N + 7) / 8, 256, 0, stream>>>(xin, batch, sums, cnts, N);
    pool_div<<<(unsigned)((gd + 255) / 256), 256, 0, stream>>>(sums, cnts, pooled, Gn);
    wmma_gemm128<<<(Gn + 15) / 16, 256, 0, stream>>>(pooled, linW, linB, (float*)d_out, Gn);
}